// QueryDepthIdentityModel_60017872994739
// MI455X (gfx1250) — compile-verified
//
#include <hip/hip_runtime.h>
#include <hip/hip_bf16.h>
#include <cstdint>

// Problem constants (from reference)
#define B_      4
#define Q_      64
#define P_      32
#define PP_     (P_*P_)          // 1024 floats per patch = 4 KB
#define H_      512
#define W_      512
#define TILE_   16
#define CHUNK_  16               // patches per LDS chunk (64 KB)
#define NCHUNK_ (Q_/CHUNK_)      // 4
#define BQ_     (B_*Q_)
#define EPS_    0.001f

// ---- CDNA5 async global->LDS copy (ASYNCcnt path), builtin if present ----
#if defined(__has_builtin)
# if __has_builtin(__builtin_amdgcn_global_load_async_to_lds_b128)
#  define USE_ASYNC_BUILTIN 1
# endif
# if __has_builtin(__builtin_amdgcn_s_wait_asynccnt)
#  define USE_WAITASYNC_BUILTIN 1
# endif
#endif

typedef int v4i_ __attribute__((vector_size(16)));           // int4 (16B)
typedef __attribute__((address_space(1))) v4i_ gv4i_;        // global int4
typedef __attribute__((address_space(3))) v4i_ lv4i_;        // LDS int4

__device__ __forceinline__ void async_copy_b128(const float* g, float* l) {
#ifdef USE_ASYNC_BUILTIN
  __builtin_amdgcn_global_load_async_to_lds_b128(
      (gv4i_*)g, (lv4i_*)l, /*offset=*/0, /*cpol=*/0);
#else
  // generic pointer to shared: low 32 bits are the LDS byte offset
  unsigned loff = (unsigned)(uintptr_t)l;
  asm volatile("global_load_async_to_lds_b128 %0, %1, off"
               :: "v"(loff), "v"(g) : "memory");
#endif
}

__device__ __forceinline__ void wait_async0() {
#ifdef USE_WAITASYNC_BUILTIN
  __builtin_amdgcn_s_wait_asynccnt(0);
#else
  asm volatile("s_wait_asynccnt 0" ::: "memory");
#endif
}

// ---------------------------------------------------------------------------
// Kernel 1: per-batch stable descending-depth sort + param reorder + mean(ds)
// ws layout (floats): [0..6)*BQ_: cx, cy, 1/sx, 1/sy, depth, ident ; 6*BQ_: src
// index (int); 7*BQ_ .. +B_: mean depth_scale per batch.
// ---------------------------------------------------------------------------
__global__ void prep_kernel(const float* __restrict__ centers,
                            const float* __restrict__ depth,
                            const float* __restrict__ sizes,
                            const float* __restrict__ depth_scale,
                            float* __restrict__ ws)
{
  const int b = blockIdx.x;
  const int i = threadIdx.x;           // 0..Q_-1
  __shared__ float dsh[Q_];
  __shared__ float ssh[Q_];
  const float di = depth[b*Q_ + i];
  dsh[i] = di;
  ssh[i] = depth_scale[b*Q_ + i];
  __syncthreads();
  // stable rank for argsort(-depth): larger depth first, ties keep index order
  int rank = 0;
  for (int j = 0; j < Q_; ++j) {
    const float dj = dsh[j];
    rank += (dj > di) || (dj == di && j < i);
  }
  const int o = b*Q_ + rank;
  ws[0*BQ_ + o] = centers[(b*Q_ + i)*2 + 0];
  ws[1*BQ_ + o] = centers[(b*Q_ + i)*2 + 1];
  ws[2*BQ_ + o] = 1.0f / sizes[(b*Q_ + i)*2 + 0];
  ws[3*BQ_ + o] = 1.0f / sizes[(b*Q_ + i)*2 + 1];
  ws[4*BQ_ + o] = di;
  ws[5*BQ_ + o] = (float)(i + 1);
  ((int*)(ws + 6*BQ_))[o] = i;
  if (i == 0) {
    float m = 0.0f;
    for (int j = 0; j < Q_; ++j) m += ssh[j];
    ws[7*BQ_ + b] = m * (1.0f/(float)Q_);
  }
}

// ---------------------------------------------------------------------------
// Kernel 2: tiled renderer. Double-buffered async-to-LDS patch staging
// (16 patches x 4KB per chunk), overlapped with the transmittance scan.
// ---------------------------------------------------------------------------
__global__ __launch_bounds__(TILE_*TILE_)
void render_kernel(const float* __restrict__ patch_alpha,
                   const float* __restrict__ ws,
                   float* __restrict__ out)
{
  extern __shared__ float smem[];
  float* buf   = smem;                        // 2 * CHUNK_ * PP_ floats (128 KB)
  float* q_cx  = smem + 2*CHUNK_*PP_;
  float* q_cy  = q_cx  + Q_;
  float* q_isx = q_cy  + Q_;
  float* q_isy = q_isx + Q_;
  float* q_dep = q_isy + Q_;
  float* q_idv = q_dep + Q_;
  int*   q_src = (int*)(q_idv + Q_);

  const int tid = threadIdx.x;                // 0..255
  const int b   = blockIdx.z;
  const int px  = blockIdx.x*TILE_ + (tid & (TILE_-1));
  const int py  = blockIdx.y*TILE_ + (tid >> 4);

  if (tid < Q_) {
    const int o = b*Q_ + tid;
    q_cx[tid]  = ws[0*BQ_ + o];
    q_cy[tid]  = ws[1*BQ_ + o];
    q_isx[tid] = ws[2*BQ_ + o];
    q_isy[tid] = ws[3*BQ_ + o];
    q_dep[tid] = ws[4*BQ_ + o];
    q_idv[tid] = ws[5*BQ_ + o];
    q_src[tid] = ((const int*)(ws + 6*BQ_))[o];
  }
  const float mean_ds = ws[7*BQ_ + b];
  __syncthreads();

  const float* pbase = patch_alpha + (size_t)b * Q_ * PP_;

  // Issue async loads for chunk 0 (256 lanes x 16B == one 4KB patch per inst)
  {
    float* dst = buf;
    for (int p = 0; p < CHUNK_; ++p) {
      const float* g = pbase + (size_t)q_src[p] * PP_ + tid*4;
      async_copy_b128(g, dst + p*PP_ + tid*4);
    }
  }

  const float gx = px * (2.0f/(float)(W_-1)) - 1.0f;
  const float gy = py * (2.0f/(float)(H_-1)) - 1.0f;

  float t = 1.0f, depth_acc = 0.0f, ident = 0.0f;
  bool hit = false;

  for (int c = 0; c < NCHUNK_; ++c) {
    wait_async0();          // this wave's chunk-c data landed in LDS
    __syncthreads();        // all waves' chunk-c data landed
    if (c + 1 < NCHUNK_) {  // overlap: prefetch next chunk during compute
      float* dst = buf + ((c+1)&1)*CHUNK_*PP_;
      for (int p = 0; p < CHUNK_; ++p) {
        const float* g = pbase + (size_t)q_src[(c+1)*CHUNK_ + p] * PP_ + tid*4;
        async_copy_b128(g, dst + p*PP_ + tid*4);
      }
    }
    const float* cbuf = buf + (c&1)*CHUNK_*PP_;
#pragma unroll 4
    for (int p = 0; p < CHUNK_; ++p) {
      const int k = c*CHUNK_ + p;
      const float lx = fminf(fmaxf((gx - q_cx[k]) * q_isx[k], -1.1f), 1.1f);
      const float ly = fminf(fmaxf((gy - q_cy[k]) * q_isy[k], -1.1f), 1.1f);
      const float ix = (lx + 1.0f) * (0.5f*(float)(P_-1));
      const float iy = (ly + 1.0f) * (0.5f*(float)(P_-1));
      const float x0f = floorf(ix), y0f = floorf(iy);
      const float wx = ix - x0f,  wy = iy - y0f;
      const int x0 = (int)x0f, y0 = (int)y0f;
      const int x1 = x0 + 1,   y1 = y0 + 1;
      const int x0c = min(max(x0,0), P_-1), x1c = min(max(x1,0), P_-1);
      const int y0c = min(max(y0,0), P_-1), y1c = min(max(y1,0), P_-1);
      const float m00 = (x0>=0 && x0<P_ && y0>=0 && y0<P_) ? 1.0f : 0.0f;
      const float m01 = (x1>=0 && x1<P_ && y0>=0 && y0<P_) ? 1.0f : 0.0f;
      const float m10 = (x0>=0 && x0<P_ && y1>=0 && y1<P_) ? 1.0f : 0.0f;
      const float m11 = (x1>=0 && x1<P_ && y1>=0 && y1<P_) ? 1.0f : 0.0f;
      const float* pat = cbuf + p*PP_;
      const float v00 = pat[y0c*P_ + x0c] * m00;   // LDS gathers
      const float v01 = pat[y0c*P_ + x1c] * m01;
      const float v10 = pat[y1c*P_ + x0c] * m10;
      const float v11 = pat[y1c*P_ + x1c] * m11;
      const float a = v00*(1.0f-wx)*(1.0f-wy) + v01*wx*(1.0f-wy)
                    + v10*(1.0f-wx)*wy        + v11*wx*wy;
      const float contrib = a * t;
      if ((contrib > EPS_) && !hit) { ident = q_idv[k]; hit = true; }
      depth_acc += contrib * q_dep[k];
      t *= (1.0f - a);
    }
    __syncthreads();        // all waves done reading cbuf before it is reused
  }

  const float occ = fminf(fmaxf(1.0f - t, 0.0f), 1.0f);
  const float vis = (occ > 0.05f) ? 1.0f : 0.0f;
  const float rid  = (hit ? ident : 0.0f) * vis;
  const float rdep = depth_acc * vis * mean_ds;

  const size_t NPIX = (size_t)B_ * H_ * W_;
  const size_t pix  = ((size_t)b*H_ + py)*W_ + px;
  out[pix]            = rid;    // rendered_identity (B,H,W)
  out[NPIX   + pix]   = rdep;   // rendered_depth    (B,1,H,W)
  out[2*NPIX + pix]   = occ;    // occ               (B,1,H,W)
}

// ---------------------------------------------------------------------------
extern "C" void kernel_launch(void* const* d_in, const int* in_sizes, int n_in,
                              void* d_out, int out_size, void* d_ws, size_t ws_size,
                              hipStream_t stream) {
  const float* patch_alpha = (const float*)d_in[0];
  const float* centers     = (const float*)d_in[1];
  const float* depth       = (const float*)d_in[2];
  const float* sizes       = (const float*)d_in[3];
  const float* depth_scale = (const float*)d_in[4];
  float* out = (float*)d_out;
  float* ws  = (float*)d_ws;   // needs (7*BQ_+B_)*4 ~ 7.2 KB

  prep_kernel<<<dim3(B_), dim3(Q_), 0, stream>>>(centers, depth, sizes,
                                                 depth_scale, ws);

  const size_t shmem = (size_t)(2*CHUNK_*PP_ + 6*Q_)*sizeof(float)
                     + (size_t)Q_*sizeof(int);   // ~130 KB < 160 KB -> 2 WG/WGP
  render_kernel<<<dim3(W_/TILE_, H_/TILE_, B_), dim3(TILE_*TILE_), shmem, stream>>>(
      patch_alpha, ws, out);
}